// BinarizeLayer_57071525429378
// MI455X (gfx1250) — compile-verified
//
#include <hip/hip_runtime.h>
#include <cstdint>
#include <cstddef>

typedef float v4f __attribute__((ext_vector_type(4)));
typedef int   v4i __attribute__((vector_size(16)));   // matches builtin V4i

#define BDIM   256
#define BTILE  16

constexpr int   Bn    = 4096;
constexpr int   Fn    = 3136;
constexpr int   Kn    = 8;
constexpr int   ROWSn = 256;
constexpr float EPSf  = 0.001f;
constexpr float Tf    = 1000.0f;

#if defined(__has_builtin)
#  if __has_builtin(__builtin_amdgcn_global_load_async_to_lds_b128)
#    define HAVE_ASYNC_LDS 1
#  endif
#endif
#ifndef HAVE_ASYNC_LDS
#  define HAVE_ASYNC_LDS 0
#endif

__global__ void zero_loss_kernel(float* loss) { *loss = 0.0f; }

__global__ __launch_bounds__(BDIM) void binarize_fused_kernel(
    const float* __restrict__ x,
    const float* __restrict__ interval,
    const float* __restrict__ i_min,
    float* __restrict__ out,
    float* __restrict__ loss)
{
    __shared__ float sRed[BDIM / 32];

#if HAVE_ASYNC_LDS
    // Stage the full 8KB interval table into LDS with the CDNA5 async path.
    // 256 threads x 32B each == ROWSn*Kn floats exactly.
    __shared__ __align__(16) float sIv[ROWSn * Kn];
    {
        auto gp = (__attribute__((address_space(1))) v4i*)(interval + threadIdx.x * 8);
        auto lp = (__attribute__((address_space(3))) v4i*)(sIv + threadIdx.x * 8);
        __builtin_amdgcn_global_load_async_to_lds_b128(gp,     lp,     0, 0);
        __builtin_amdgcn_global_load_async_to_lds_b128(gp + 1, lp + 1, 0, 0);
        asm volatile("s_wait_asynccnt 0" ::: "memory");
    }
    __syncthreads();
    const float* ivtab = sIv;
#else
    const float* ivtab = interval;
#endif

    const int  tid   = threadIdx.x;
    const int  f     = blockIdx.x * BDIM + tid;
    const int  bb    = blockIdx.y * BTILE;
    const bool valid = (f < Fn);

    float acc = 0.0f;

    if (valid) {
        // Per-feature bin centers: clamped cumsum of interval row + i_min.
        const int row = (f < 3072) ? (f >> 4) : (f - 2880);
        const float* ivp = ivtab + row * Kn;
        float c[Kn];
        float run = i_min[f];
#pragma unroll
        for (int k = 0; k < Kn; ++k) {
            float w = ivp[k];
            w = (w > EPSf) ? w : EPSf;
            run += w;
            c[k] = run;
        }

        const float* xr = x + (size_t)bb * Fn + f;
        float*       op = out + ((size_t)bb * Fn + (size_t)f) * Kn;

        for (int i = 0; i < BTILE; ++i) {
            float xv = __builtin_nontemporal_load(xr);
            if (i + 2 < BTILE) __builtin_prefetch(xr + 2 * (size_t)Fn, 0, 0);

            float d[Kn];
#pragma unroll
            for (int k = 0; k < Kn; ++k) { float t = xv - c[k]; d[k] = t * t; }

            // first-min argmin == jnp.argmax of softmax (first tie)
            float dmin = d[0]; int amin = 0;
#pragma unroll
            for (int k = 1; k < Kn; ++k)
                if (d[k] < dmin) { dmin = d[k]; amin = k; }

            float e[Kn]; float sum = 0.0f;
#pragma unroll
            for (int k = 0; k < Kn; ++k) {
                e[k] = __expf(-Tf * (d[k] - dmin));
                sum += e[k];
            }
            const float inv = 1.0f / sum;

            v4f v0, v1;
            float li = 0.0f;
#pragma unroll
            for (int k = 0; k < Kn; ++k) {
                float s  = e[k] * inv;
                li      += d[k] * s;
                float oh = (k == amin) ? 1.0f : 0.0f;
                float v  = (oh - s) + s;   // straight-through forward value
                if (k < 4) v0[k] = v; else v1[k - 4] = v;
            }
            acc += li;

            // Streaming output (411MB, write-once): non-temporal hints.
            __builtin_nontemporal_store(v0, (v4f*)op);
            __builtin_nontemporal_store(v1, (v4f*)(op + 4));

            xr += Fn;
            op += (size_t)Fn * Kn;
        }
    }

    // Block-level loss reduction: wave32 shfl tree -> LDS -> one atomic/block.
#pragma unroll
    for (int off = 16; off > 0; off >>= 1)
        acc += __shfl_xor(acc, off, 32);
    if ((tid & 31) == 0) sRed[tid >> 5] = acc;
    __syncthreads();
    if (tid == 0) {
        float t = 0.0f;
#pragma unroll
        for (int w = 0; w < BDIM / 32; ++w) t += sRed[w];
        atomicAdd(loss, t * (1.0f / (float)Bn));
    }
}

extern "C" void kernel_launch(void* const* d_in, const int* in_sizes, int n_in,
                              void* d_out, int out_size, void* d_ws, size_t ws_size,
                              hipStream_t stream) {
    const float* x        = (const float*)d_in[0];
    const float* interval = (const float*)d_in[1];
    const float* i_min    = (const float*)d_in[2];
    float* out  = (float*)d_out;
    float* loss = out + (size_t)Bn * Fn * Kn;   // kmeans_loss scalar after total_out

    zero_loss_kernel<<<1, 1, 0, stream>>>(loss);

    dim3 grid((Fn + BDIM - 1) / BDIM, Bn / BTILE, 1);
    binarize_fused_kernel<<<grid, dim3(BDIM, 1, 1), 0, stream>>>(
        x, interval, i_min, out, loss);

    (void)in_sizes; (void)n_in; (void)out_size; (void)d_ws; (void)ws_size;
}